// SeqAttnMatch_66400194396534
// MI455X (gfx1250) — compile-verified
//
#include <hip/hip_runtime.h>
#include <hip/hip_bf16.h>
#include <math.h>

typedef __attribute__((ext_vector_type(2))) float v2f;
typedef __attribute__((ext_vector_type(4))) float v4f;
typedef __attribute__((ext_vector_type(8))) float v8f;

#define NB   64
#define NL1  1024
#define NL2  1024
#define NH   128
#define NEG_BIG (-1e30f)

static __device__ __forceinline__ v8f wmma_f32(v2f a, v2f b, v8f c) {
  // D = A(16x4) * B(4x16) + C(16x16), fp32
  return __builtin_amdgcn_wmma_f32_16x16x4_f32(
      /*neg_a=*/false, a, /*neg_b=*/false, b,
      /*c_mod=*/(short)0, c, /*reuse_a=*/false, /*reuse_b=*/false);
}

// ---------------------------------------------------------------------------
// Kernel 1: P = relu(X @ W^T + b) for x (first 512 blocks) and y (last 512).
// Workgroup = 256 threads = 8 waves; each wave -> 16 rows x 128 cols of output.
// W is staged in LDS, XOR-swizzled in k-pairs so B-fragment b64 reads avoid
// bank conflicts.
// ---------------------------------------------------------------------------
__global__ __launch_bounds__(256) void proj_kernel(
    const float* __restrict__ x, const float* __restrict__ y,
    const float* __restrict__ W, const float* __restrict__ bias,
    float* __restrict__ xp, float* __restrict__ yp) {
  __shared__ __align__(16) float lds_w[NH * NH];  // 64 KB, swizzled

  const int tid = threadIdx.x;
  // Load + swizzle W: element W[h][k] -> lds_w[h*128 + ((k>>1)^(h&31))*2 + (k&1)]
  #pragma unroll
  for (int it = 0; it < 64; ++it) {
    int idx = tid + 256 * it;           // 16384 elements
    int h = idx >> 7, k = idx & 127;
    lds_w[h * 128 + (((k >> 1) ^ (h & 31)) << 1) + (k & 1)] = W[idx];
  }
  __syncthreads();

  const int blk = blockIdx.x;           // 0..1023
  const float* src;
  float* dst;
  int rowbase;
  if (blk < 512) { src = x;  dst = xp; rowbase = blk * 128; }
  else           { src = y;  dst = yp; rowbase = (blk - 512) * 128; }

  const int wave = tid >> 5, lane = tid & 31;
  const int lm = lane & 15, lh = lane >> 4;   // lane-in-half, half
  const int nbase = rowbase + wave * 16;

  v8f acc[8];
  #pragma unroll
  for (int t = 0; t < 8; ++t) {
    float bv = bias[t * 16 + lm];
    #pragma unroll
    for (int c = 0; c < 8; ++c) acc[t][c] = bv;
  }

  const float* arow = src + (size_t)(nbase + lm) * NH;
  #pragma unroll
  for (int kc = 0; kc < 32; ++kc) {
    const int ke = kc * 4 + 2 * lh;           // even
    v2f a = *(const v2f*)(arow + ke);
    #pragma unroll
    for (int t = 0; t < 8; ++t) {
      const int h = t * 16 + lm;
      v2f bf = *(const v2f*)(&lds_w[h * 128 + (((ke >> 1) ^ (h & 31)) << 1)]);
      acc[t] = wmma_f32(a, bf, acc[t]);
    }
  }

  // relu + store (D layout: VGPR r = rows r / r+8, N = lane%16)
  #pragma unroll
  for (int t = 0; t < 8; ++t)
    #pragma unroll
    for (int r = 0; r < 8; ++r) {
      float v = acc[t][r];
      v = v > 0.0f ? v : 0.0f;
      dst[(size_t)(nbase + r + 8 * lh) * NH + t * 16 + lm] = v;
    }
}

// ---------------------------------------------------------------------------
// Kernel 2: fused flash-style attention.
// grid = (L1/128, B); workgroup = 256 threads = 8 waves; wave w owns rows
// ibase = iblk*128 + w*16. Online softmax over j in chunks of 32.
// ---------------------------------------------------------------------------
__global__ __launch_bounds__(256) void attn_kernel(
    const float* __restrict__ xp, const float* __restrict__ ypj,
    const float* __restrict__ y, const unsigned char* __restrict__ ymask,
    float* __restrict__ out) {
  __shared__ __align__(16) float s_yp[32 * 132];       // y_proj chunk, padded
  __shared__ __align__(16) float s_yt[NH * 36];        // y chunk transposed [h][j]
  __shared__ __align__(16) float s_p[8 * 16 * 36];     // per-wave P scratch [i][j]
  __shared__ float s_m[32];                            // mask chunk (1=pad)

  const int b = blockIdx.y, iblk = blockIdx.x;
  const int tid = threadIdx.x, wave = tid >> 5, lane = tid & 31;
  const int lm = lane & 15, lh = lane >> 4;
  const int ibase = iblk * 128 + wave * 16;

  // Preload A fragments of x_proj for all K=128 (32 chunks of 4)
  const float* xprow = xp + ((size_t)b * NL1 + ibase + lm) * NH;
  v2f afrag[32];
  #pragma unroll
  for (int kc = 0; kc < 32; ++kc) afrag[kc] = *(const v2f*)(xprow + kc * 4 + 2 * lh);

  v8f O[8];
  float m[8], l[8];
  #pragma unroll
  for (int t = 0; t < 8; ++t)
    #pragma unroll
    for (int c = 0; c < 8; ++c) O[t][c] = 0.0f;
  #pragma unroll
  for (int r = 0; r < 8; ++r) { m[r] = -INFINITY; l[r] = 0.0f; }

  const float* ypb = ypj + (size_t)b * NL2 * NH;
  const float* yb  = y   + (size_t)b * NL2 * NH;
  const unsigned char* mb = ymask + (size_t)b * NL2;
  float* pw = &s_p[wave * 16 * 36];

  for (int j0 = 0; j0 < NL2; j0 += 32) {
    __syncthreads();  // protect LDS stages from previous iteration's readers

    // Stage y_proj[b, j0:j0+32, :] -> s_yp [32][132]
    #pragma unroll
    for (int it = 0; it < 4; ++it) {
      int idx = tid + 256 * it;          // 1024 float4 slots
      int j = idx >> 5, c4 = idx & 31;
      v4f v = *(const v4f*)(ypb + (size_t)(j0 + j) * NH + c4 * 4);
      *(v4f*)(&s_yp[j * 132 + c4 * 4]) = v;
    }
    // Stage y[b, j0:j0+32, :] transposed -> s_yt[h][j], stride 36
    #pragma unroll
    for (int it = 0; it < 16; ++it) {
      int idx = tid + 256 * it;          // 4096 floats
      int j = idx >> 7, h = idx & 127;
      s_yt[h * 36 + j] = yb[(size_t)(j0 + j) * NH + h];
    }
    if (tid < 32) s_m[tid] = mb[j0 + tid] ? 1.0f : 0.0f;
    __syncthreads();

    // S tiles: [16 i x 32 j] = two 16x16 WMMA outputs, K = H = 128
    v8f S[2];
    #pragma unroll
    for (int jt = 0; jt < 2; ++jt)
      #pragma unroll
      for (int c = 0; c < 8; ++c) S[jt][c] = 0.0f;

    #pragma unroll
    for (int kc = 0; kc < 32; ++kc) {
      const int ke = kc * 4 + 2 * lh;
      v2f a = afrag[kc];
      #pragma unroll
      for (int jt = 0; jt < 2; ++jt) {
        v2f bf = *(const v2f*)(&s_yp[(jt * 16 + lm) * 132 + ke]);
        S[jt] = wmma_f32(a, bf, S[jt]);
      }
    }

    // Mask (each lane's whole v8f shares one j column)
    #pragma unroll
    for (int jt = 0; jt < 2; ++jt) {
      float mk = s_m[jt * 16 + lm];
      #pragma unroll
      for (int c = 0; c < 8; ++c)
        S[jt][c] = (mk > 0.5f) ? NEG_BIG : S[jt][c];
    }

    // Online softmax: row r lives in component r across the 16 lanes of a half
    float mnew[8], scale[8];
    #pragma unroll
    for (int r = 0; r < 8; ++r) {
      float t = fmaxf(S[0][r], S[1][r]);
      t = fmaxf(t, __shfl_xor(t, 1, 32));
      t = fmaxf(t, __shfl_xor(t, 2, 32));
      t = fmaxf(t, __shfl_xor(t, 4, 32));
      t = fmaxf(t, __shfl_xor(t, 8, 32));
      mnew[r] = fmaxf(m[r], t);
      scale[r] = expf(m[r] - mnew[r]);   // exp(-inf)=0 on first chunk
    }
    #pragma unroll
    for (int r = 0; r < 8; ++r) {
      S[0][r] = expf(S[0][r] - mnew[r]);
      S[1][r] = expf(S[1][r] - mnew[r]);
      float rs = S[0][r] + S[1][r];
      rs += __shfl_xor(rs, 1, 32);
      rs += __shfl_xor(rs, 2, 32);
      rs += __shfl_xor(rs, 4, 32);
      rs += __shfl_xor(rs, 8, 32);
      l[r] = l[r] * scale[r] + rs;
      m[r] = mnew[r];
    }
    #pragma unroll
    for (int t = 0; t < 8; ++t)
      #pragma unroll
      for (int c = 0; c < 8; ++c) O[t][c] *= scale[c];

    // C-layout -> A-layout via per-wave LDS scratch: pw[i][j], stride 36
    #pragma unroll
    for (int jt = 0; jt < 2; ++jt)
      #pragma unroll
      for (int r = 0; r < 8; ++r)
        pw[(r + 8 * lh) * 36 + jt * 16 + lm] = S[jt][r];
    __syncthreads();  // lockstep across waves; guarantees P visibility

    // O += P(16x32) * Y(32x128)
    #pragma unroll
    for (int kc = 0; kc < 8; ++kc) {
      const int ke = kc * 4 + 2 * lh;   // K = j offset within chunk
      v2f pa = *(const v2f*)(&pw[lm * 36 + ke]);
      #pragma unroll
      for (int t = 0; t < 8; ++t) {
        v2f bf = *(const v2f*)(&s_yt[(t * 16 + lm) * 36 + ke]);
        O[t] = wmma_f32(pa, bf, O[t]);
      }
    }
  }

  // Normalize and store
  float inv[8];
  #pragma unroll
  for (int r = 0; r < 8; ++r) inv[r] = 1.0f / l[r];
  float* orow = out + ((size_t)b * NL1 + ibase) * NH;
  #pragma unroll
  for (int t = 0; t < 8; ++t)
    #pragma unroll
    for (int r = 0; r < 8; ++r)
      orow[(size_t)(r + 8 * lh) * NH + t * 16 + lm] = O[t][r] * inv[r];
}

// ---------------------------------------------------------------------------
extern "C" void kernel_launch(void* const* d_in, const int* in_sizes, int n_in,
                              void* d_out, int out_size, void* d_ws, size_t ws_size,
                              hipStream_t stream) {
  const float* x = (const float*)d_in[0];                 // [B, L1, H]
  const float* y = (const float*)d_in[1];                 // [B, L2, H]
  const unsigned char* ymask = (const unsigned char*)d_in[2];  // [B, L2] bool
  const float* W = (const float*)d_in[3];                 // [H, H]
  const float* bias = (const float*)d_in[4];              // [H]
  float* out = (float*)d_out;                             // [B, L1, H]

  float* xp = (float*)d_ws;                               // [B*L1, H]
  float* yp = xp + (size_t)NB * NL1 * NH;                 // [B*L2, H]

  // 1024 blocks: 512 for x rows, 512 for y rows; 128 rows per block
  proj_kernel<<<1024, 256, 0, stream>>>(x, y, W, bias, xp, yp);
  // (L1/128) x B blocks
  attn_kernel<<<dim3(NL1 / 128, NB), 256, 0, stream>>>(xp, yp, y, ymask, out);
}